// GINBackbone_41669772706621
// MI455X (gfx1250) — compile-verified
//
#include <hip/hip_runtime.h>

// GIN backbone for MI455X (gfx1250, wave32).
// Layers: z = h + scatter_add(h[src] -> dst); h' = relu(relu(z@W1+b1)@W2+b2)
// Final: global mean pool over sorted batch ids.
// Sizes: N=100000 nodes, E=600000 edges, H=128, L=6, G=1024.

#define GIN_N 100000
#define GIN_E 600000
#define GIN_H 128
#define GIN_L 6
#define GIN_G 1024

typedef __attribute__((ext_vector_type(2))) float v2f;
typedef __attribute__((ext_vector_type(8))) float v8f;
typedef __attribute__((ext_vector_type(4))) unsigned int u32x4;
typedef __attribute__((ext_vector_type(8))) int i32x8;
typedef __attribute__((ext_vector_type(4))) int i32x4;

#if defined(__has_builtin)
#if __has_builtin(__builtin_amdgcn_tensor_load_to_lds) && \
    __has_builtin(__builtin_amdgcn_s_wait_tensorcnt)
#define USE_TDM 1
#endif
#endif
#ifndef USE_TDM
#define USE_TDM 0
#endif

// ---------------------------------------------------------------- copy h -> z
__global__ __launch_bounds__(256) void copy_f4_kernel(float4* __restrict__ dst,
                                                      const float4* __restrict__ src,
                                                      int n4) {
  int i = blockIdx.x * 256 + threadIdx.x;
  int stride = gridDim.x * 256;
  for (; i < n4; i += stride) dst[i] = src[i];
}

// ------------------------------------------------- scatter-add  z[dst] += h[src]
// One lane handles 4 consecutive features of one edge (32 lanes = 128 feats/edge).
// Both h and z are L2-resident (51.2 MB each vs 192 MB L2): f32 atomics at L2.
__global__ __launch_bounds__(256) void scatter_add_kernel(float* __restrict__ z,
                                                          const float* __restrict__ h,
                                                          const int* __restrict__ srcI,
                                                          const int* __restrict__ dstI) {
  int idx = blockIdx.x * 256 + threadIdx.x;       // E*32 = 19.2M work items
  if (idx >= GIN_E * 32) return;
  int e = idx >> 5;
  int q = (idx & 31) * 4;
  int s = srcI[e];
  int d = dstI[e];
  const float4 v = *(const float4*)(h + (size_t)s * GIN_H + q);
  float* p = z + (size_t)d * GIN_H + q;
  atomicAdd(p + 0, v.x);
  atomicAdd(p + 1, v.y);
  atomicAdd(p + 2, v.z);
  atomicAdd(p + 3, v.w);
}

// ---------------------------------------------------- fused GEMM + bias + ReLU
// Out[r,:] = relu(A[r,:] @ W + bias), A: nRows x 128, W: 128 x 128.
// Block = 256 threads = 8 waves; block tile = 128 rows x 128 cols.
// Wave tile = 32 rows x 64 cols (2 row-tiles x 4 col-tiles, 8 WMMA accs).
// A-panel: TDM tensor_load_to_lds with pad_enable -> padded stride 132 floats,
//          tensor_dim1 = nRows-m0 -> hardware zero-fill for the tail block.
// W      : staged transposed (sWt[col*132+k]) so each B fragment is one aligned
//          ds_load_b64; stride 132 => (4*col+kb)%64 banks, conflict-free.
// In-place (Out==A) safe: all A reads (TDM, pre-barrier) precede stores.
__global__ __launch_bounds__(256) void gemm_relu_kernel(const float* __restrict__ A,
                                                        const float* __restrict__ W,
                                                        const float* __restrict__ bias,
                                                        float* __restrict__ Out,
                                                        int nRows) {
  __shared__ float sA[128 * 132];
  __shared__ float sWt[128 * 132];

  const int tid = threadIdx.x;
  const int m0 = blockIdx.x * 128;

#if USE_TDM
  if (tid < 32) {  // wave 0 issues the tensor DMA (EXEC all-ones in this wave)
    const unsigned long long ga =
        (unsigned long long)(uintptr_t)(A + (size_t)m0 * GIN_H);
    const unsigned int lds_off = (unsigned int)(uintptr_t)(void*)sA;
    const unsigned int tdim1 = (unsigned int)(nRows - m0);  // rows available

    u32x4 g0;
    g0[0] = 1u;                                   // count=1, user mode
    g0[1] = lds_off;                              // lds_addr (bytes)
    g0[2] = (unsigned int)(ga & 0xffffffffull);   // global_addr[31:0]
    g0[3] = (unsigned int)((ga >> 32) & 0x1ffffffull) | (2u << 30);  // type=2

    i32x8 g1;
    g1[0] = (int)((2u << 16)      // data_size = 4B
                  | (1u << 20)    // pad_enable
                  | (6u << 22)    // pad_interval: 128 dwords (one 512B row)
                  | (3u << 25));  // pad_amount: 4 dwords -> LDS stride 132
    g1[1] = (int)(128u << 16);                    // tensor_dim0 = 128
    g1[2] = (int)((tdim1 & 0xffffu) << 16);       // tensor_dim1[15:0]
    g1[3] = (int)(((tdim1 >> 16) & 0xffffu)       // tensor_dim1[31:16]
                  | (128u << 16));                // tile_dim0 = 128
    g1[4] = 128;                                  // tile_dim1 = 128 (tile_dim2=0)
    g1[5] = 128;                                  // tensor_dim0_stride = 128
    g1[6] = 0;
    g1[7] = 0;

    i32x4 gz4 = {0, 0, 0, 0};
    i32x8 gz8 = {0, 0, 0, 0, 0, 0, 0, 0};
    __builtin_amdgcn_tensor_load_to_lds(g0, g1, gz4, gz4, gz8, 0);
  }
#endif

  // All threads: stage W transposed into padded LDS (one-time, 16K floats).
#pragma unroll 4
  for (int j = 0; j < 64; ++j) {
    int idx = j * 256 + tid;
    int k = idx >> 7;
    int c = idx & 127;
    sWt[c * 132 + k] = W[idx];
  }

#if USE_TDM
  if (tid < 32) __builtin_amdgcn_s_wait_tensorcnt(0);
#else
  // Fallback: cooperative A staging with zero-fill for tail rows.
#pragma unroll 4
  for (int j = 0; j < 64; ++j) {
    int idx = j * 256 + tid;
    int r = idx >> 7;
    int c = idx & 127;
    int row = m0 + r;
    sA[r * 132 + c] = (row < nRows) ? A[(size_t)row * GIN_H + c] : 0.f;
  }
#endif
  __syncthreads();

  const int lane = tid & 31;
  const int laneM = lane & 15;
  const int half = lane >> 4;        // lane-half selects K pair / M offset
  const int wave = tid >> 5;
  const int rowG = (wave >> 1) * 32; // 0,32,64,96
  const int colG = (wave & 1) * 64;  // 0,64

  v8f acc[2][4] = {};
  float bv[4];
#pragma unroll
  for (int t = 0; t < 4; ++t) bv[t] = bias[colG + t * 16 + laneM];

  const float* __restrict__ pA = &sA[(rowG + laneM) * 132];
  const float* __restrict__ pB = &sWt[(colG + laneM) * 132];

#pragma unroll 2
  for (int k0 = 0; k0 < 128; k0 += 4) {
    const int kb = k0 + half * 2;
    const v2f a0 = *(const v2f*)(pA + kb);             // rows rowG..+15
    const v2f a1 = *(const v2f*)(pA + 16 * 132 + kb);  // rows rowG+16..+31
#pragma unroll
    for (int t = 0; t < 4; ++t) {
      const v2f b = *(const v2f*)(pB + t * 16 * 132 + kb);
      acc[0][t] = __builtin_amdgcn_wmma_f32_16x16x4_f32(
          false, a0, false, b, (short)0, acc[0][t], false, false);
      acc[1][t] = __builtin_amdgcn_wmma_f32_16x16x4_f32(
          false, a1, false, b, (short)0, acc[1][t], false, false);
    }
  }

  // Epilogue. C/D layout: VGPR v -> row (+half*8+v), lane -> col.
  if (m0 + 128 <= nRows) {  // fast path: 781 of 782 blocks, no predication
#pragma unroll
    for (int rt = 0; rt < 2; ++rt) {
#pragma unroll
      for (int t = 0; t < 4; ++t) {
        const int col = colG + t * 16 + laneM;
#pragma unroll
        for (int v = 0; v < 8; ++v) {
          const int row = m0 + rowG + rt * 16 + half * 8 + v;
          float x = acc[rt][t][v] + bv[t];
          Out[(size_t)row * GIN_H + col] = x > 0.f ? x : 0.f;
        }
      }
    }
  } else {
#pragma unroll
    for (int rt = 0; rt < 2; ++rt) {
#pragma unroll
      for (int t = 0; t < 4; ++t) {
        const int col = colG + t * 16 + laneM;
#pragma unroll
        for (int v = 0; v < 8; ++v) {
          const int row = m0 + rowG + rt * 16 + half * 8 + v;
          if (row < nRows) {
            float x = acc[rt][t][v] + bv[t];
            Out[(size_t)row * GIN_H + col] = x > 0.f ? x : 0.f;
          }
        }
      }
    }
  }
}

// ------------------------------------------------------------------- pooling
__global__ __launch_bounds__(256) void zero_kernel(float* __restrict__ p, int n) {
  int i = blockIdx.x * 256 + threadIdx.x;
  if (i < n) p[i] = 0.f;
}

__global__ __launch_bounds__(256) void pool_acc_kernel(const float* __restrict__ x,
                                                       const int* __restrict__ batch,
                                                       float* __restrict__ gsum,
                                                       float* __restrict__ counts) {
  int idx = blockIdx.x * 256 + threadIdx.x;       // N*32 work items
  if (idx >= GIN_N * 32) return;
  int n = idx >> 5;
  int q = (idx & 31) * 4;
  int b = batch[n];
  const float4 v = *(const float4*)(x + (size_t)n * GIN_H + q);
  float* p = gsum + (size_t)b * GIN_H + q;
  atomicAdd(p + 0, v.x);
  atomicAdd(p + 1, v.y);
  atomicAdd(p + 2, v.z);
  atomicAdd(p + 3, v.w);
  if (q == 0) atomicAdd(counts + b, 1.0f);
}

__global__ __launch_bounds__(256) void pool_div_kernel(float* __restrict__ g,
                                                       const float* __restrict__ counts) {
  int i = blockIdx.x * 256 + threadIdx.x;         // G*H
  if (i >= GIN_G * GIN_H) return;
  float c = counts[i >> 7];
  c = c < 1.f ? 1.f : c;
  g[i] = g[i] / c;
}

// -------------------------------------------------------------------- driver
extern "C" void kernel_launch(void* const* d_in, const int* in_sizes, int n_in,
                              void* d_out, int out_size, void* d_ws, size_t ws_size,
                              hipStream_t stream) {
  const float* x     = (const float*)d_in[0];
  const int*   ei    = (const int*)d_in[1];
  const int*   batch = (const int*)d_in[2];
  const float* W1    = (const float*)d_in[3];
  const float* b1    = (const float*)d_in[4];
  const float* W2    = (const float*)d_in[5];
  const float* b2    = (const float*)d_in[6];

  const int* srcI = ei;
  const int* dstI = ei + GIN_E;

  float* xout = (float*)d_out;                       // N*H
  float* g    = xout + (size_t)GIN_N * GIN_H;        // G*H

  float* Wbuf0  = (float*)d_ws;                      // N*H
  float* Wbuf1  = Wbuf0 + (size_t)GIN_N * GIN_H;     // N*H
  float* counts = Wbuf1 + (size_t)GIN_N * GIN_H;     // G

  const int nh4      = GIN_N * GIN_H / 4;            // 3.2M float4
  const int copyBlks = 2048;                         // grid-stride
  const int scatBlks = (GIN_E * 32 + 255) / 256;
  const int gemmBlks = (GIN_N + 127) / 128;
  const int poolBlks = (GIN_N * 32 + 255) / 256;
  const int zeroN    = GIN_G * GIN_H;

  const float* hin = x;
  for (int l = 0; l < GIN_L; ++l) {
    float* work = (l & 1) ? Wbuf1 : Wbuf0;
    float* out  = (l == GIN_L - 1) ? xout : work;
    const float* w1  = W1 + (size_t)l * GIN_H * GIN_H;
    const float* w2  = W2 + (size_t)l * GIN_H * GIN_H;
    const float* bb1 = b1 + (size_t)l * GIN_H;
    const float* bb2 = b2 + (size_t)l * GIN_H;

    copy_f4_kernel<<<copyBlks, 256, 0, stream>>>((float4*)work, (const float4*)hin, nh4);
    scatter_add_kernel<<<scatBlks, 256, 0, stream>>>(work, hin, srcI, dstI);
    gemm_relu_kernel<<<gemmBlks, 256, 0, stream>>>(work, w1, bb1, work, GIN_N); // in-place
    gemm_relu_kernel<<<gemmBlks, 256, 0, stream>>>(work, w2, bb2, out, GIN_N);
    hin = out;
  }

  // global mean pool
  zero_kernel<<<(zeroN + 255) / 256, 256, 0, stream>>>(g, zeroN);
  zero_kernel<<<(GIN_G + 255) / 256, 256, 0, stream>>>(counts, GIN_G);
  pool_acc_kernel<<<poolBlks, 256, 0, stream>>>(xout, batch, g, counts);
  pool_div_kernel<<<(GIN_G * GIN_H + 255) / 256, 256, 0, stream>>>(g, counts);
}